// multi_view_depth_loss_6468220748287
// MI455X (gfx1250) — compile-verified
//
#include <hip/hip_runtime.h>

#define BB 8
#define VV 3
#define HH 512
#define WW 640
#define HWW (HH * WW)

// workspace layout in floats
#define WS_KINV 0      // 8 * 16
#define WS_RTINV 128   // 24 * 16
#define WS_P 512       // 72 * 16  (indexed [(b*9 + pair)*16 + r*4 + c])
#define WS_ACC 1664    // num[9], den[9]

typedef float v2f __attribute__((ext_vector_type(2)));
typedef float v8f __attribute__((ext_vector_type(8)));

// ---------------------------------------------------------------------------
// General 4x4 inverse via adjugate (gluInvertMatrix form; works for row- or
// column-major flat arrays identically since inv(M^T) = inv(M)^T).
// ---------------------------------------------------------------------------
__device__ inline void inv4(const float* __restrict__ m, float* __restrict__ o) {
    float inv[16];
    inv[0]  =  m[5]*m[10]*m[15] - m[5]*m[11]*m[14] - m[9]*m[6]*m[15] + m[9]*m[7]*m[14] + m[13]*m[6]*m[11] - m[13]*m[7]*m[10];
    inv[4]  = -m[4]*m[10]*m[15] + m[4]*m[11]*m[14] + m[8]*m[6]*m[15] - m[8]*m[7]*m[14] - m[12]*m[6]*m[11] + m[12]*m[7]*m[10];
    inv[8]  =  m[4]*m[9]*m[15]  - m[4]*m[11]*m[13] - m[8]*m[5]*m[15] + m[8]*m[7]*m[13] + m[12]*m[5]*m[11] - m[12]*m[7]*m[9];
    inv[12] = -m[4]*m[9]*m[14]  + m[4]*m[10]*m[13] + m[8]*m[5]*m[14] - m[8]*m[6]*m[13] - m[12]*m[5]*m[10] + m[12]*m[6]*m[9];
    inv[1]  = -m[1]*m[10]*m[15] + m[1]*m[11]*m[14] + m[9]*m[2]*m[15] - m[9]*m[3]*m[14] - m[13]*m[2]*m[11] + m[13]*m[3]*m[10];
    inv[5]  =  m[0]*m[10]*m[15] - m[0]*m[11]*m[14] - m[8]*m[2]*m[15] + m[8]*m[3]*m[14] + m[12]*m[2]*m[11] - m[12]*m[3]*m[10];
    inv[9]  = -m[0]*m[9]*m[15]  + m[0]*m[11]*m[13] + m[8]*m[1]*m[15] - m[8]*m[3]*m[13] - m[12]*m[1]*m[11] + m[12]*m[3]*m[9];
    inv[13] =  m[0]*m[9]*m[14]  - m[0]*m[10]*m[13] - m[8]*m[1]*m[14] + m[8]*m[2]*m[13] + m[12]*m[1]*m[10] - m[12]*m[2]*m[9];
    inv[2]  =  m[1]*m[6]*m[15]  - m[1]*m[7]*m[14]  - m[5]*m[2]*m[15] + m[5]*m[3]*m[14] + m[13]*m[2]*m[7]  - m[13]*m[3]*m[6];
    inv[6]  = -m[0]*m[6]*m[15]  + m[0]*m[7]*m[14]  + m[4]*m[2]*m[15] - m[4]*m[3]*m[14] - m[12]*m[2]*m[7]  + m[12]*m[3]*m[6];
    inv[10] =  m[0]*m[5]*m[15]  - m[0]*m[7]*m[13]  - m[4]*m[1]*m[15] + m[4]*m[3]*m[13] + m[12]*m[1]*m[7]  - m[12]*m[3]*m[5];
    inv[14] = -m[0]*m[5]*m[14]  + m[0]*m[6]*m[13]  + m[4]*m[1]*m[14] - m[4]*m[2]*m[13] - m[12]*m[1]*m[6]  + m[12]*m[2]*m[5];
    inv[3]  = -m[1]*m[6]*m[11]  + m[1]*m[7]*m[10]  + m[5]*m[2]*m[11] - m[5]*m[3]*m[10] - m[9]*m[2]*m[7]   + m[9]*m[3]*m[6];
    inv[7]  =  m[0]*m[6]*m[11]  - m[0]*m[7]*m[10]  - m[4]*m[2]*m[11] + m[4]*m[3]*m[10] + m[8]*m[2]*m[7]   - m[8]*m[3]*m[6];
    inv[11] = -m[0]*m[5]*m[11]  + m[0]*m[7]*m[9]   + m[4]*m[1]*m[11] - m[4]*m[3]*m[9]  - m[8]*m[1]*m[7]   + m[8]*m[3]*m[5];
    inv[15] =  m[0]*m[5]*m[10]  - m[0]*m[6]*m[9]   - m[4]*m[1]*m[10] + m[4]*m[2]*m[9]  + m[8]*m[1]*m[6]   - m[8]*m[2]*m[5];
    float det = m[0]*inv[0] + m[1]*inv[4] + m[2]*inv[8] + m[3]*inv[12];
    float id = 1.0f / det;
#pragma unroll
    for (int k = 0; k < 16; ++k) o[k] = inv[k] * id;
}

// ---------------------------------------------------------------------------
// Kernel 1: analytic inverses of K (8) and RT (24); zero the accumulators.
// ---------------------------------------------------------------------------
__global__ __launch_bounds__(32) void mvdl_inv_kernel(const float* __restrict__ K,
                                                      const float* __restrict__ RT,
                                                      float* __restrict__ ws) {
    int t = threadIdx.x;
    if (t < BB) inv4(K + t * 16, ws + WS_KINV + t * 16);
    int m = t - BB;
    if (m >= 0 && m < BB * VV) inv4(RT + m * 16, ws + WS_RTINV + m * 16);
    if (t < 18) ws[WS_ACC + t] = 0.0f;
}

// ---------------------------------------------------------------------------
// Kernel 2: P[b][i][j] = K * RT_j * RTinv_i * Kinv via V_WMMA_F32_16X16X4_F32.
// One wave per (pair, batch-half); 4 independent 4x4 products per WMMA as the
// diagonal blocks of (16x4)x(4x16). LDS restages C-layout -> A-layout.
//   A layout: lanes 0-15: M=lane, VGPR0=K0, VGPR1=K1; lanes 16-31: K2,K3.
//   B layout: VGPRr: lanes 0-15 -> K=r, lanes 16-31 -> K=r+2; N=lane&15.
//   C layout: VGPRr: lanes 0-15 -> M=r, lanes 16-31 -> M=r+8; N=lane&15.
// ---------------------------------------------------------------------------
__global__ __launch_bounds__(32) void mvdl_chain_kernel(const float* __restrict__ K,
                                                        const float* __restrict__ RT,
                                                        float* __restrict__ ws) {
    __shared__ float tile[16][17];
    const float* Kinv  = ws + WS_KINV;
    const float* RTinv = ws + WS_RTINV;
    float* P = ws + WS_P;

    int g    = blockIdx.x;   // 0..17
    int p    = g >> 1;       // pair 0..8
    int hb   = g & 1;        // batch half (batches 4*hb .. 4*hb+3)
    int i    = p / 3;
    int j    = p % 3;
    int lane = threadIdx.x;
    int half = lane >> 4;    // K/M half select
    int idx  = lane & 15;    // A row / B,C column
    int q    = idx >> 2;     // 4x4 block index
    int r    = idx & 3;      // row (A) / col (B) within block
    int bq   = 4 * hb + q;   // batch handled by this lane's block

    v8f cz = {0.f, 0.f, 0.f, 0.f, 0.f, 0.f, 0.f, 0.f};
    v2f a, bv;

    // ---- T1 = K * RT_j  (4 batches batched block-diagonally) ----
    a.x  = K[bq * 16 + r * 4 + 2 * half];
    a.y  = K[bq * 16 + r * 4 + 2 * half + 1];
    bv.x = RT[(bq * VV + j) * 16 + (2 * half + 0) * 4 + r];
    bv.y = RT[(bq * VV + j) * 16 + (2 * half + 1) * 4 + r];
    v8f c1 = __builtin_amdgcn_wmma_f32_16x16x4_f32(false, a, false, bv,
                                                   (short)0, cz, false, false);
#pragma unroll
    for (int rr = 0; rr < 8; ++rr) tile[rr + 8 * half][idx] = c1[rr];
    __syncthreads();

    // ---- T2 = T1 * RTinv_i ----
    a.x  = tile[idx][(idx & 12) + 2 * half];
    a.y  = tile[idx][(idx & 12) + 2 * half + 1];
    bv.x = RTinv[(bq * VV + i) * 16 + (2 * half + 0) * 4 + r];
    bv.y = RTinv[(bq * VV + i) * 16 + (2 * half + 1) * 4 + r];
    __syncthreads();
    v8f c2 = __builtin_amdgcn_wmma_f32_16x16x4_f32(false, a, false, bv,
                                                   (short)0, cz, false, false);
#pragma unroll
    for (int rr = 0; rr < 8; ++rr) tile[rr + 8 * half][idx] = c2[rr];
    __syncthreads();

    // ---- P = T2 * Kinv ----
    a.x  = tile[idx][(idx & 12) + 2 * half];
    a.y  = tile[idx][(idx & 12) + 2 * half + 1];
    bv.x = Kinv[bq * 16 + (2 * half + 0) * 4 + r];
    bv.y = Kinv[bq * 16 + (2 * half + 1) * 4 + r];
    v8f c3 = __builtin_amdgcn_wmma_f32_16x16x4_f32(false, a, false, bv,
                                                   (short)0, cz, false, false);

    // scatter diagonal blocks: element (M, N=idx) kept when M>>2 == idx>>2
#pragma unroll
    for (int rr = 0; rr < 8; ++rr) {
        int M = rr + 8 * half;
        if ((M >> 2) == q) {
            P[(bq * 9 + p) * 16 + (M & 3) * 4 + r] = c3[rr];
        }
    }
}

// ---------------------------------------------------------------------------
// Kernel 3: per-pixel reprojection + bilinear sample + masked L1 reduction.
// grid = (HW/256, B, 9 pairs). P is uniform per block; gathers hit L2
// (whole pred_pts = 31.5 MB << 192 MB L2).
// ---------------------------------------------------------------------------
__global__ __launch_bounds__(256) void mvdl_main_kernel(const float* __restrict__ pred,
                                                        const float* __restrict__ Pw,
                                                        float* __restrict__ accum) {
    int b = blockIdx.y;
    int p = blockIdx.z;
    int i = p / 3;
    int j = p % 3;
    const float* Pm = Pw + (b * 9 + p) * 16;
    float P00 = Pm[0],  P01 = Pm[1],  P02 = Pm[2],  P03 = Pm[3];
    float P10 = Pm[4],  P11 = Pm[5],  P12 = Pm[6],  P13 = Pm[7];
    float P20 = Pm[8],  P21 = Pm[9],  P22 = Pm[10], P23 = Pm[11];
    float P30 = Pm[12], P31 = Pm[13], P32 = Pm[14], P33 = Pm[15];

    int pix = blockIdx.x * 256 + threadIdx.x;   // grid sized exactly to HW
    int x = pix % WW;
    int y = pix / WW;
    float fx = (float)x, fy = (float)y;

    float d = pred[(b * VV + i) * HWW + pix];

    float X  = d * (fx * P00 + fy * P01 + P02) + P03;
    float Y  = d * (fx * P10 + fy * P11 + P12) + P13;
    float Z  = d * (fx * P20 + fy * P21 + P22) + P23;
    float Wc = d * (fx * P30 + fy * P31 + P32) + P33;

    float px = X / Wc, py = Y / Wc, pd = Z / Wc;
    float gx = px * (2.0f / (float)(WW - 1)) - 1.0f;
    float gy = py * (2.0f / (float)(HH - 1)) - 1.0f;

    float num = 0.0f, den = 0.0f;
    if (gx >= -1.0f && gx <= 1.0f && gy >= -1.0f && gy <= 1.0f) {
        const float* img = pred + (b * VV + j) * HWW;
        float ix = ((gx + 1.0f) * (float)WW - 1.0f) * 0.5f;
        float iy = ((gy + 1.0f) * (float)HH - 1.0f) * 0.5f;
        float x0f = floorf(ix), y0f = floorf(iy);
        float wx1 = ix - x0f,  wy1 = iy - y0f;
        int x0 = (int)x0f, y0 = (int)y0f;
        int x1 = x0 + 1,   y1 = y0 + 1;
        int x0c = min(max(x0, 0), WW - 1), x1c = min(max(x1, 0), WW - 1);
        int y0c = min(max(y0, 0), HH - 1), y1c = min(max(y1, 0), HH - 1);
        float v00 = img[y0c * WW + x0c];
        float v10 = img[y0c * WW + x1c];
        float v01 = img[y1c * WW + x0c];
        float v11 = img[y1c * WW + x1c];
        v00 = (x0 >= 0 && x0 < WW && y0 >= 0 && y0 < HH) ? v00 : 0.0f;
        v10 = (x1 >= 0 && x1 < WW && y0 >= 0 && y0 < HH) ? v10 : 0.0f;
        v01 = (x0 >= 0 && x0 < WW && y1 >= 0 && y1 < HH) ? v01 : 0.0f;
        v11 = (x1 >= 0 && x1 < WW && y1 >= 0 && y1 < HH) ? v11 : 0.0f;
        float warped = (1.0f - wy1) * (1.0f - wx1) * v00 + (1.0f - wy1) * wx1 * v10
                     + wy1 * (1.0f - wx1) * v01 + wy1 * wx1 * v11;
        num = fabsf(warped - pd);
        den = 1.0f;
    }

    // wave32 reduction, then cross-wave via LDS, 2 atomics per block
#pragma unroll
    for (int off = 16; off > 0; off >>= 1) {
        num += __shfl_down(num, off, 32);
        den += __shfl_down(den, off, 32);
    }
    __shared__ float sn[8], sd[8];
    int wid = threadIdx.x >> 5, lid = threadIdx.x & 31;
    if (lid == 0) { sn[wid] = num; sd[wid] = den; }
    __syncthreads();
    if (threadIdx.x == 0) {
        float tn = 0.0f, td = 0.0f;
#pragma unroll
        for (int w = 0; w < 8; ++w) { tn += sn[w]; td += sd[w]; }
        atomicAdd(&accum[p], tn);
        atomicAdd(&accum[9 + p], td);
    }
}

// ---------------------------------------------------------------------------
// Kernel 4: total = sum_p num[p] / max(den[p], 1)
// ---------------------------------------------------------------------------
__global__ void mvdl_finalize_kernel(const float* __restrict__ accum,
                                     float* __restrict__ out) {
    if (threadIdx.x == 0 && blockIdx.x == 0) {
        float t = 0.0f;
#pragma unroll
        for (int p = 0; p < 9; ++p) t += accum[p] / fmaxf(accum[9 + p], 1.0f);
        out[0] = t;
    }
}

extern "C" void kernel_launch(void* const* d_in, const int* in_sizes, int n_in,
                              void* d_out, int out_size, void* d_ws, size_t ws_size,
                              hipStream_t stream) {
    (void)in_sizes; (void)n_in; (void)out_size; (void)ws_size;
    const float* pred = (const float*)d_in[0];
    const float* Kc   = (const float*)d_in[1];
    const float* RT   = (const float*)d_in[2];
    float* ws = (float*)d_ws;

    mvdl_inv_kernel<<<1, 32, 0, stream>>>(Kc, RT, ws);
    mvdl_chain_kernel<<<18, 32, 0, stream>>>(Kc, RT, ws);
    dim3 grid(HWW / 256, BB, 9);
    mvdl_main_kernel<<<grid, 256, 0, stream>>>(pred, ws + WS_P, ws + WS_ACC);
    mvdl_finalize_kernel<<<1, 32, 0, stream>>>(ws + WS_ACC, (float*)d_out);
}